// RelPosMHAXLChunked_24704651887080
// MI455X (gfx1250) — compile-verified
//
#include <hip/hip_runtime.h>
#include <hip/hip_bf16.h>
#include <math.h>

// ---------------------------------------------------------------------------
// RelPos MHA-XL (Transformer-XL attention), fp32 WMMA on gfx1250 (wave32).
//   B=8, L=512, E=1024, H=16, HD=64, SCALE=1/sqrt(E)=1/32
// ---------------------------------------------------------------------------

typedef float v2f __attribute__((ext_vector_type(2)));
typedef float v8f __attribute__((ext_vector_type(8)));

#define BATCH 8
#define SEQL  512
#define EMB   1024
#define NH    16
#define HD3   192          // 3*HD per head in qkv raw layout
#define SCALE 0.03125f     // 1/sqrt(1024)

// D = A(16x4) * B(4x16) + C, fp32 exact.
// VGPR layouts (ISA 7.12.2):
//   A: lane<16 -> row=lane, K={0,1}; lane>=16 -> row=lane-16, K={2,3}
//   B: lane<16 -> col=lane, K={0,1}; lane>=16 -> col=lane-16, K={2,3}
//   C/D: col=lane%16, rows = (lane>=16 ? 8:0)+v  for v in 0..7
__device__ __forceinline__ v8f wmma4(v2f a, v2f b, v8f c) {
  return __builtin_amdgcn_wmma_f32_16x16x4_f32(false, a, false, b, (short)0, c,
                                               false, false);
}

// ---------------------------------------------------------------------------
// Generic C[M,N] = A[M,K] @ Bw[N,K]^T (+bias).  One wave -> 16x64 tile.
// Software-pipelined K loop: iteration k issues the loads for group k+4
// BEFORE the WMMAs for group k, so ~5 b64 loads are always in flight while
// the matrix pipe runs.  Ragged M handled by clamping the A row (loads stay
// in-bounds, result row discarded) + uniform fast path vs guarded tail store.
// ---------------------------------------------------------------------------
__global__ void gemm_xt_kernel(const float* __restrict__ A,
                               const float* __restrict__ Bw,
                               const float* __restrict__ bias,
                               float* __restrict__ C,
                               int M, int N, int K, int lda, int ldb, int ldc) {
  const int wave = (blockIdx.x * blockDim.x + threadIdx.x) >> 5;
  const int lane = threadIdx.x & 31;
  const int half = lane >> 4;
  const int ln   = lane & 15;
  const int ntiles = N >> 6;                 // 64-wide N tiles
  const int tm = wave / ntiles;
  const int tn = (wave - tm * ntiles) << 6;
  const int m0 = tm << 4;
  if (m0 >= M) return;                       // wave-uniform

  int m = m0 + ln;
  if (m > M - 1) m = M - 1;                  // clamp: load valid, discard later
  const float* arow = A + (size_t)m * lda + (half << 1);

  // Hoisted B row pointers: per-iteration address = bp[t] + k0 (imm offset).
  const float* bp[4];
#pragma unroll
  for (int t = 0; t < 4; ++t)
    bp[t] = Bw + (size_t)(tn + (t << 4) + ln) * ldb + (half << 1);

  v8f acc[4];
#pragma unroll
  for (int t = 0; t < 4; ++t) acc[t] = v8f{};

  // ---- software pipeline: prologue loads group 0 ----
  v2f afc = {arow[0], arow[1]};
  v2f bfc[4];
#pragma unroll
  for (int t = 0; t < 4; ++t) {
    bfc[t].x = bp[t][0];
    bfc[t].y = bp[t][1];
  }

#pragma unroll 2
  for (int k0 = 0; k0 < K - 4; k0 += 4) {
    // issue next group's loads first (overlap with WMMAs below)
    v2f afn = {arow[k0 + 4], arow[k0 + 5]};
    v2f bfn[4];
#pragma unroll
    for (int t = 0; t < 4; ++t) {
      bfn[t].x = bp[t][k0 + 4];
      bfn[t].y = bp[t][k0 + 5];
    }
    // compute current group
#pragma unroll
    for (int t = 0; t < 4; ++t)
      acc[t] = wmma4(afc, bfc[t], acc[t]);
    afc = afn;
#pragma unroll
    for (int t = 0; t < 4; ++t) bfc[t] = bfn[t];
  }
  // epilogue: last group
#pragma unroll
  for (int t = 0; t < 4; ++t)
    acc[t] = wmma4(afc, bfc[t], acc[t]);

  if (m0 + 16 <= M) {                        // wave-uniform: full tile
#pragma unroll
    for (int t = 0; t < 4; ++t) {
      const int n = tn + (t << 4) + ln;
      const float bb = bias ? bias[n] : 0.0f;
#pragma unroll
      for (int v = 0; v < 8; ++v) {
        const int mm = m0 + (half << 3) + v;
        C[(size_t)mm * ldc + n] = acc[t][v] + bb;
      }
    }
  } else {                                   // ragged tail tile
#pragma unroll
    for (int t = 0; t < 4; ++t) {
      const int n = tn + (t << 4) + ln;
      const float bb = bias ? bias[n] : 0.0f;
#pragma unroll
      for (int v = 0; v < 8; ++v) {
        const int mm = m0 + (half << 3) + v;
        if (mm < M) C[(size_t)mm * ldc + n] = acc[t][v] + bb;
      }
    }
  }
}

// ---------------------------------------------------------------------------
// Fused scores: attn_raw[b,h,q,k] = (q+bu)*S . k   +  (q+bv)*S . p_k[k-q+511]
// grid = (4 k-chunks of 128, 32 q-strips of 16, B*H), block = 32 (one wave).
// BD computed as a 16x144 Toeplitz band via WMMA into LDS, read shifted
// (band col = kk + 15 - m  <=>  p = k - q + 511).
// Band column 143 is the only one whose pos index can be out of range (1023),
// and it is never read (max read col = 142) -> pointer clamp, no predication.
// ---------------------------------------------------------------------------
__global__ void scores_kernel(const float* __restrict__ qkv,   // (4096,3072)
                              const float* __restrict__ pk,    // (1023,1024)
                              const float* __restrict__ bu,    // flat 1024
                              const float* __restrict__ bv,    // flat 1024
                              float* __restrict__ attn) {      // (B,H,L,L)
  const int k0 = blockIdx.x << 7;            // k-chunk start (0,128,256,384)
  const int q0 = blockIdx.y << 4;            // q-strip start
  const int bh = blockIdx.z;                 // b*16+h
  const int b  = bh >> 4, h = bh & 15;
  const int lane = threadIdx.x;
  const int half = lane >> 4;
  const int ln   = lane & 15;
  const int pbase = 496 - q0 + k0;           // band col 0 -> pos index pbase

  __shared__ float band[16][144];            // 9,216 B

  // Preload A fragments (q_u, q_v) for full K=64 head dim: 16 steps of K=4.
  const float* qrow =
      qkv + ((size_t)(b * SEQL + q0 + ln)) * (3 * EMB) + h * HD3 + (half << 1);
  const float* buh = bu + h * 64 + (half << 1);
  const float* bvh = bv + h * 64 + (half << 1);
  v2f aU[16], aV[16];
#pragma unroll
  for (int s = 0; s < 16; ++s) {
    const int k = s << 2;
    const float x0 = qrow[k], x1 = qrow[k + 1];
    aU[s].x = (x0 + buh[k]) * SCALE;  aU[s].y = (x1 + buh[k + 1]) * SCALE;
    aV[s].x = (x0 + bvh[k]) * SCALE;  aV[s].y = (x1 + bvh[k + 1]) * SCALE;
  }

  // BD band: 9 column tiles of 16 (cols 0..143, col c -> pos pbase+c)
  for (int t = 0; t < 9; ++t) {
    int pcol = pbase + (t << 4) + ln;
    if (pcol > 1022) pcol = 1022;            // only col 143 (never read)
    const float* pkc = pk + (size_t)pcol * EMB + h * 64 + (half << 1);
    v2f bf[16];
#pragma unroll
    for (int s = 0; s < 16; ++s) {           // load phase
      bf[s].x = pkc[s << 2];
      bf[s].y = pkc[(s << 2) + 1];
    }
    v8f acc = {};
#pragma unroll
    for (int s = 0; s < 16; ++s)             // compute phase
      acc = wmma4(aV[s], bf[s], acc);
#pragma unroll
    for (int v = 0; v < 8; ++v)
      band[(half << 3) + v][(t << 4) + ln] = acc[v];
  }
  __syncthreads();                           // single wave; LDS ordering

  // AC tiles (8 x 16 cols) + shifted-band combine, write raw scores.
  for (int t = 0; t < 8; ++t) {
    const int kcol = k0 + (t << 4) + ln;     // key index 0..511
    const float* krow = qkv + ((size_t)(b * SEQL + kcol)) * (3 * EMB) +
                        h * HD3 + 64 + (half << 1);
    v2f bf[16];
#pragma unroll
    for (int s = 0; s < 16; ++s) {
      bf[s].x = krow[s << 2];
      bf[s].y = krow[(s << 2) + 1];
    }
    v8f acc = {};
#pragma unroll
    for (int s = 0; s < 16; ++s)
      acc = wmma4(aU[s], bf[s], acc);
#pragma unroll
    for (int v = 0; v < 8; ++v) {
      const int m  = (half << 3) + v;        // row within strip
      const int kk = (t << 4) + ln;          // col within 128-chunk
      const float bd = band[m][kk + 15 - m]; // shift: p = k - q + 511
      const size_t idx =
          (((size_t)bh * SEQL) + q0 + m) * SEQL + k0 + kk;
      attn[idx] = acc[v] + bd;
    }
  }
}

// ---------------------------------------------------------------------------
// In-place softmax over rows of 512. One wave per row (wave32, 16 elem/lane).
// ---------------------------------------------------------------------------
__global__ void softmax_kernel(float* __restrict__ attn) {
  const int row  = (blockIdx.x * blockDim.x + threadIdx.x) >> 5;
  const int lane = threadIdx.x & 31;
  float* p = attn + (size_t)row * SEQL;
  float vals[16];
  float mx = -INFINITY;
#pragma unroll
  for (int i = 0; i < 16; ++i) {
    vals[i] = p[lane + (i << 5)];
    mx = fmaxf(mx, vals[i]);
  }
#pragma unroll
  for (int off = 16; off > 0; off >>= 1) mx = fmaxf(mx, __shfl_xor(mx, off, 32));
  float sum = 0.0f;
#pragma unroll
  for (int i = 0; i < 16; ++i) {
    vals[i] = __expf(vals[i] - mx);
    sum += vals[i];
  }
#pragma unroll
  for (int off = 16; off > 0; off >>= 1) sum += __shfl_xor(sum, off, 32);
  const float inv = 1.0f / sum;
#pragma unroll
  for (int i = 0; i < 16; ++i) p[lane + (i << 5)] = vals[i] * inv;
}

// ---------------------------------------------------------------------------
// ctx[b, q, h*64+d] = sum_k attn[b,h,q,k] * v[b,k,h,d]
// grid = (32 q-strips, H, B), block = 32 (one wave -> 16x64 tile).
// Software-pipelined like the GEMM: next group's loads precede this group's
// WMMAs.
// ---------------------------------------------------------------------------
__global__ void av_kernel(const float* __restrict__ attn,
                          const float* __restrict__ qkv,
                          float* __restrict__ ctx) {
  const int q0 = blockIdx.x << 4;
  const int h  = blockIdx.y;
  const int b  = blockIdx.z;
  const int lane = threadIdx.x;
  const int half = lane >> 4;
  const int ln   = lane & 15;

  const float* arow =
      attn + (((size_t)(b * NH + h) * SEQL) + q0 + ln) * SEQL + (half << 1);
  const float* vbase = qkv + ((size_t)(b * SEQL + (half << 1))) * (3 * EMB) +
                       h * HD3 + 128 + ln;

  v8f acc[4];
#pragma unroll
  for (int t = 0; t < 4; ++t) acc[t] = v8f{};

  // prologue: group 0
  v2f afc = {arow[0], arow[1]};
  v2f bfc[4];
#pragma unroll
  for (int t = 0; t < 4; ++t) {
    bfc[t].x = vbase[t * 16];
    bfc[t].y = vbase[3 * EMB + t * 16];
  }

#pragma unroll 2
  for (int kb = 0; kb < SEQL - 4; kb += 4) {
    v2f afn = {arow[kb + 4], arow[kb + 5]};
    const float* vr = vbase + (size_t)(kb + 4) * (3 * EMB);
    v2f bfn[4];
#pragma unroll
    for (int t = 0; t < 4; ++t) {
      bfn[t].x = vr[t * 16];
      bfn[t].y = vr[3 * EMB + t * 16];
    }
#pragma unroll
    for (int t = 0; t < 4; ++t)
      acc[t] = wmma4(afc, bfc[t], acc[t]);
    afc = afn;
#pragma unroll
    for (int t = 0; t < 4; ++t) bfc[t] = bfn[t];
  }
#pragma unroll
  for (int t = 0; t < 4; ++t)
    acc[t] = wmma4(afc, bfc[t], acc[t]);

#pragma unroll
  for (int t = 0; t < 4; ++t) {
#pragma unroll
    for (int v = 0; v < 8; ++v) {
      const int m = (half << 3) + v;
      const int n = (t << 4) + ln;
      ctx[((size_t)(b * SEQL + q0 + m)) * EMB + h * 64 + n] = acc[t][v];
    }
  }
}

// ---------------------------------------------------------------------------
extern "C" void kernel_launch(void* const* d_in, const int* in_sizes, int n_in,
                              void* d_out, int out_size, void* d_ws,
                              size_t ws_size, hipStream_t stream) {
  const float* query    = (const float*)d_in[0];
  const float* pos_embs = (const float*)d_in[3];
  const float* w_in     = (const float*)d_in[4];   // (3072,1024)
  const float* w_pos    = (const float*)d_in[5];   // (1024,1024)
  const float* bu       = (const float*)d_in[6];   // flat 1024
  const float* bv       = (const float*)d_in[7];   // flat 1024
  const float* w_out    = (const float*)d_in[8];   // (1024,1024)
  const float* b_out    = (const float*)d_in[9];   // (1024)

  float* out  = (float*)d_out;                                  // (8,512,1024)
  float* attn = (float*)d_out + (size_t)BATCH * SEQL * EMB;     // (8,16,512,512)

  float* qkv = (float*)d_ws;                                    // (4096,3072)
  float* pk  = qkv + (size_t)BATCH * SEQL * 3 * EMB;            // (1023,1024)
  float* ctx = pk + (size_t)1024 * EMB;                         // (4096,1024)

  const int M  = BATCH * SEQL;   // 4096
  const int MP = 2 * SEQL - 1;   // 1023

  // 1) qkv = query @ w_in^T : waves = (4096/16)*(3072/64) = 256*48 = 12288
  gemm_xt_kernel<<<12288 / 4, 128, 0, stream>>>(query, w_in, nullptr, qkv,
                                                M, 3 * EMB, EMB,
                                                EMB, EMB, 3 * EMB);
  // 2) pk = pos_embs @ w_pos^T : waves = ceil(1023/16)*(1024/64) = 64*16 = 1024
  gemm_xt_kernel<<<1024 / 4, 128, 0, stream>>>(pos_embs, w_pos, nullptr, pk,
                                               MP, EMB, EMB,
                                               EMB, EMB, EMB);
  // 3) raw scores (AC + shifted BD) -> attn region
  scores_kernel<<<dim3(4, 32, BATCH * NH), 32, 0, stream>>>(qkv, pk, bu, bv,
                                                            attn);
  // 4) softmax in place: 8*16*512 = 65536 rows, 4 waves / block
  softmax_kernel<<<65536 / 4, 128, 0, stream>>>(attn);
  // 5) ctx = attn @ v
  av_kernel<<<dim3(32, NH, BATCH), 32, 0, stream>>>(attn, qkv, ctx);
  // 6) out = ctx @ w_out^T + b_out : waves = 256*(1024/64) = 4096
  gemm_xt_kernel<<<4096 / 4, 128, 0, stream>>>(ctx, w_out, b_out, out,
                                               M, EMB, EMB,
                                               EMB, EMB, EMB);
}